// ExpertGroupRouter_30039001268734
// MI455X (gfx1250) — compile-verified
//
#include <hip/hip_runtime.h>
#include <math.h>

typedef __attribute__((ext_vector_type(2))) float v2f;
typedef __attribute__((ext_vector_type(4))) float v4f;
typedef __attribute__((ext_vector_type(8))) float v8f;

#define DDIM 2048
#define NEXP 16
#define KCHUNK 128
#define LSTRIDE 132           // 128 + 4 pad: keeps b64 alignment (even), spreads banks
#define WAVES 4
#define TOK_PER_BLOCK 64      // 4 waves * 16 tokens
#define SC_STRIDE 20
#define NTOK (4 * 4096)
#define MAXE 6
#define THRV 0.15f

__global__ __launch_bounds__(128) void router_main(
    const float* __restrict__ x, const float* __restrict__ We,
    const float* __restrict__ Wg, float* __restrict__ wOut,
    float* __restrict__ iOut, int* __restrict__ counts)
{
    __shared__ float xs[WAVES * 16 * LSTRIDE];   // per-wave 16-token x tiles
    __shared__ float wes[16 * LSTRIDE];          // W_expert chunk (16 x KCHUNK)
    __shared__ float wgs[16 * LSTRIDE];          // W_group chunk, rows 2..15 stay zero
    __shared__ float sc[WAVES * 16 * SC_STRIDE]; // per-token scores (16 expert + 2 group)
    __shared__ int   hist[NEXP];

    const int tid  = threadIdx.x;
    const int lane = tid & 31;
    const int wave = tid >> 5;
    const int half = lane >> 4;  // lane half selects K offset (+2)
    const int l16  = lane & 15;

    // zero group-weight region once (rows 2..15 act as zero B-columns) + histogram
    for (int i = tid; i < 16 * LSTRIDE; i += 128) wgs[i] = 0.0f;
    if (tid < NEXP) hist[tid] = 0;

    const int waveTok = blockIdx.x * TOK_PER_BLOCK + wave * 16;

    v8f ce = {};  // 16 tokens x 16 experts
    v8f cg = {};  // 16 tokens x [g0, g1, 0...]

    for (int k0 = 0; k0 < DDIM; k0 += KCHUNK) {
        __syncthreads();
        // ---- stage this wave's 16-token x tile (512 float4s, 16 per lane) ----
        {
            float* xw = xs + wave * 16 * LSTRIDE;
            #pragma unroll
            for (int i = 0; i < 16; ++i) {
                int idx = lane + 32 * i;       // 0..511
                int row = idx >> 5, c4 = idx & 31;
                v4f v = *(const v4f*)(x + (size_t)(waveTok + row) * DDIM + k0 + c4 * 4);
                float* dst = xw + row * LSTRIDE + c4 * 4;
                v2f lo = {v.x, v.y}, hi = {v.z, v.w};
                *(v2f*)dst = lo;
                *(v2f*)(dst + 2) = hi;
            }
        }
        // ---- stage W_expert chunk block-wide (512 float4s, 4 per thread) ----
        #pragma unroll
        for (int i = 0; i < 4; ++i) {
            int idx = tid + 128 * i;
            int row = idx >> 5, c4 = idx & 31;
            v4f v = *(const v4f*)(We + (size_t)row * DDIM + k0 + c4 * 4);
            float* dst = wes + row * LSTRIDE + c4 * 4;
            v2f lo = {v.x, v.y}, hi = {v.z, v.w};
            *(v2f*)dst = lo;
            *(v2f*)(dst + 2) = hi;
        }
        // ---- stage W_group rows 0..1 (64 float4s) ----
        if (tid < 64) {
            int row = tid >> 5, c4 = tid & 31;
            v4f v = *(const v4f*)(Wg + (size_t)row * DDIM + k0 + c4 * 4);
            float* dst = wgs + row * LSTRIDE + c4 * 4;
            v2f lo = {v.x, v.y}, hi = {v.z, v.w};
            *(v2f*)dst = lo;
            *(v2f*)(dst + 2) = hi;
        }
        __syncthreads();

        // ---- WMMA K loop: A 16x4 (tokens x k), B 4x16 (k x experts/groups) ----
        const float* xw = xs + wave * 16 * LSTRIDE;
        #pragma unroll
        for (int kk = 0; kk < KCHUNK; kk += 4) {
            int off = kk + 2 * half;
            v2f a  = *(const v2f*)(xw  + l16 * LSTRIDE + off);
            v2f be = *(const v2f*)(wes + l16 * LSTRIDE + off);
            v2f bg = *(const v2f*)(wgs + l16 * LSTRIDE + off);
            ce = __builtin_amdgcn_wmma_f32_16x16x4_f32(false, a, false, be,
                                                       (short)0, ce, false, false);
            cg = __builtin_amdgcn_wmma_f32_16x16x4_f32(false, a, false, bg,
                                                       (short)0, cg, false, false);
        }
    }

    // ---- spill scores to LDS: C layout vgpr i, lane L -> M=i+8*half, N=l16 ----
    {
        float* scw = sc + wave * 16 * SC_STRIDE;
        #pragma unroll
        for (int i = 0; i < 8; ++i) {
            int m = i + 8 * half;
            scw[m * SC_STRIDE + l16] = ce[i];
            if (l16 < 2) scw[m * SC_STRIDE + 16 + l16] = cg[i];
        }
    }

    // ---- per-token routing: lanes 0..15, one token each ----
    if (lane < 16) {
        const float* s = sc + (wave * 16 + lane) * SC_STRIDE;
        float sv[16];
        #pragma unroll
        for (int e = 0; e < NEXP; ++e) sv[e] = s[e];
        float g0 = 1.0f / (1.0f + expf(-s[16]));
        float g1 = 1.0f / (1.0f + expf(-s[17]));

        // group A: experts 0..7, top1 softmax prob
        float mA = sv[0]; int aIdx = 0;
        #pragma unroll
        for (int e = 1; e < 8; ++e) if (sv[e] > mA) { mA = sv[e]; aIdx = e; }
        float dA = 0.0f;
        #pragma unroll
        for (int e = 0; e < 8; ++e) dA += expf(sv[e] - mA);
        float aBest = 1.0f / dA;

        // group B: experts 8..11, top1 softmax prob * sigmoid gate * threshold
        float mB = sv[8]; int bIdx = 8;
        #pragma unroll
        for (int e = 9; e < 12; ++e) if (sv[e] > mB) { mB = sv[e]; bIdx = e; }
        float dB = 0.0f;
        #pragma unroll
        for (int e = 8; e < 12; ++e) dB += expf(sv[e] - mB);
        float bw = (g0 > THRV) ? (1.0f / dB) * g0 : 0.0f;

        // group C: experts 12..15, softmax, top-2 by raw score (first-occurrence ties)
        float svc[4];
        #pragma unroll
        for (int j = 0; j < 4; ++j) svc[j] = sv[12 + j];
        float mC = svc[0];
        #pragma unroll
        for (int j = 1; j < 4; ++j) mC = fmaxf(mC, svc[j]);
        float pc[4]; float dC = 0.0f;
        #pragma unroll
        for (int j = 0; j < 4; ++j) { pc[j] = expf(svc[j] - mC); dC += pc[j]; }
        float invC = 1.0f / dC;
        int i1 = 0;
        #pragma unroll
        for (int j = 1; j < 4; ++j) if (svc[j] > svc[i1]) i1 = j;
        int i2 = (i1 == 0) ? 1 : 0;
        #pragma unroll
        for (int j = 0; j < 4; ++j) {
            if (j != i1 && svc[j] > svc[i2]) i2 = j;
        }
        float actC = (g1 > THRV) ? g1 : 0.0f;
        float cw0 = pc[i1] * invC * actC;
        float cw1 = pc[i2] * invC * actC;

        float sum = aBest + bw + cw0 + cw1;
        float inv = 1.0f / (sum + 1e-8f);
        float w[MAXE] = {aBest * inv, bw * inv, cw0 * inv, cw1 * inv, 0.0f, 0.0f};
        int  id[MAXE] = {aIdx, bIdx, 12 + i1, 12 + i2, 0, 0};

        int tok = waveTok + lane;
        #pragma unroll
        for (int j = 0; j < MAXE; ++j) {
            wOut[(size_t)tok * MAXE + j] = w[j];
            iOut[(size_t)tok * MAXE + j] = (float)id[j];
            atomicAdd(&hist[id[j]], 1);
        }
    }

    __syncthreads();
    if (tid < NEXP) atomicAdd(&counts[tid], hist[tid]);
}

__global__ void zero_counts(int* counts) {
    if (threadIdx.x < NEXP) counts[threadIdx.x] = 0;
}

__global__ void aux_kernel(const int* __restrict__ counts, float* __restrict__ out) {
    if (threadIdx.x == 0) {
        float total = 0.0f, c[NEXP];
        for (int e = 0; e < NEXP; ++e) { c[e] = (float)counts[e]; total += c[e]; }
        const float u = 1.0f / (float)NEXP;
        const float lu = logf(u);
        float aux = 0.0f;
        for (int e = 0; e < NEXP; ++e) aux += u * (lu - logf(c[e] / total));
        out[0] = aux * 0.01f;
    }
}

extern "C" void kernel_launch(void* const* d_in, const int* in_sizes, int n_in,
                              void* d_out, int out_size, void* d_ws, size_t ws_size,
                              hipStream_t stream) {
    const float* x  = (const float*)d_in[0];   // (4, 4096, 2048)
    const float* We = (const float*)d_in[1];   // (16, 2048)
    const float* Wg = (const float*)d_in[2];   // (2, 2048)
    float* wOut = (float*)d_out;               // weights: NTOK*6
    float* iOut = wOut + (size_t)NTOK * MAXE;  // indices (as float): NTOK*6
    float* aux  = wOut + 2 * (size_t)NTOK * MAXE; // scalar aux
    int* counts = (int*)d_ws;

    zero_counts<<<1, 32, 0, stream>>>(counts);
    router_main<<<NTOK / TOK_PER_BLOCK, 128, 0, stream>>>(x, We, Wg, wOut, iOut, counts);
    aux_kernel<<<1, 32, 0, stream>>>(counts, aux);
}